// SupervisedGraphSage_49039936585979
// MI455X (gfx1250) — compile-verified
//
#include <hip/hip_runtime.h>
#include <hip/hip_bf16.h>

typedef float v2f __attribute__((ext_vector_type(2)));
typedef float v4f __attribute__((ext_vector_type(4)));
typedef float v8f __attribute__((ext_vector_type(8)));

#define FDIM 256      // feature dim F
#define KDIM 512      // 2F (concat width / GEMM K)
#define DEG  32       // fixed neighbor count
#define TM   16       // output rows per block (one WMMA M tile)
#define THREADS 128   // 4 waves of 32

// Fully fused GraphSAGE forward for a 16-row strip of the batch:
//   S = sum_d concat(feat_data[nb], feats[nb])        (vectorized b128 gather)
//   T = concat(feat_data[nodes], S @ W1^T + D*b1)     (WMMA GEMM1, exact fp32)
//   U = T @ W2^T + b2 ; out = U / max(||U||_2, 1e-12) (WMMA GEMM2 + row norm)
__global__ __launch_bounds__(THREADS)
void graphsage_fused_kernel(const float* __restrict__ feat_data,
                            const float* __restrict__ feats,
                            const float* __restrict__ W1,
                            const float* __restrict__ b1,
                            const float* __restrict__ W2,
                            const float* __restrict__ b2,
                            const int*   __restrict__ adj,
                            const int*   __restrict__ nodes,
                            int nNodes,
                            float* __restrict__ out)
{
    __shared__ float S[TM][KDIM];     // gathered neighbor sums (cols 0..255 reused for U later)
    __shared__ float T[TM][KDIM];     // concat(node_feats, neighbor_emb)
    __shared__ int   nidx[TM];
    __shared__ int   nbr[TM][DEG];
    __shared__ float part[TM][8];
    __shared__ float inv_norm[TM];

    const int t    = threadIdx.x;
    const int m0   = blockIdx.x * TM;
    const int lane = t & 31;
    const int wave = t >> 5;

    // ---- Phase 0: cache node ids and adjacency rows in LDS ----
    if (t < TM) {
        int g = m0 + t;
        nidx[t] = (g < nNodes) ? nodes[g] : 0;   // pad rows read node 0 (never written out)
    }
    __syncthreads();
    for (int i = t; i < TM * DEG; i += THREADS)
        nbr[i >> 5][i & 31] = adj[(size_t)nidx[i >> 5] * DEG + (i & 31)];
    __syncthreads();

    // ---- Phase 1: gather + neighbor sum, b128 loads (16B per lane) ----
    // 128 threads = 2 half-rows x 64 lanes x 4 consecutive floats.
    {
        const int half = t >> 6;          // which of two rows this thread serves
        const int tc   = (t & 63) << 2;   // column base 0..252, multiple of 4 (16B aligned)
        for (int r = 0; r < TM; r += 2) {
            const int rr  = r + half;
            const int idx = nidx[rr];
            float s0 = 0.f, s1 = 0.f, s2 = 0.f, s3 = 0.f;   // sum feat_data[nb]
            float f0 = 0.f, f1 = 0.f, f2 = 0.f, f3 = 0.f;   // sum feats[nb]
            #pragma unroll 8
            for (int d = 0; d < DEG; ++d) {
                const int j = nbr[rr][d];
                const v4f fa = *(const v4f*)(feat_data + (size_t)j * FDIM + tc);
                const v4f fb = *(const v4f*)(feats     + (size_t)j * FDIM + tc);
                s0 += fa.x; s1 += fa.y; s2 += fa.z; s3 += fa.w;
                f0 += fb.x; f1 += fb.y; f2 += fb.z; f3 += fb.w;
            }
            v4f sv; sv.x = s0; sv.y = s1; sv.z = s2; sv.w = s3;
            v4f fv; fv.x = f0; fv.y = f1; fv.z = f2; fv.w = f3;
            *(v4f*)(&S[rr][tc])        = sv;
            *(v4f*)(&S[rr][FDIM + tc]) = fv;
            *(v4f*)(&T[rr][tc]) = *(const v4f*)(feat_data + (size_t)idx * FDIM + tc);
        }
    }
    __syncthreads();

    const int lr    = lane & 15;          // row-in-tile (A) / col-in-tile (B, C/D)
    const int kh    = (lane >> 4) << 1;   // K sub-offset: 0 or 2
    const int rbase = (lane >> 4) << 3;   // C/D row base: 0 or 8

    // ---- Phase 2: neighbor_emb = S @ W1^T + DEG*b1  -> T[:, 256:512] ----
    {
        v8f acc0 = {}, acc1 = {}, acc2 = {}, acc3 = {};
        for (int k0 = 0; k0 < KDIM; k0 += 4) {
            v2f a;
            a.x = S[lr][k0 + kh];
            a.y = S[lr][k0 + kh + 1];
            const int nb0 = wave * 64 + lr;
            const float* w0 = W1 + (size_t)(nb0     ) * KDIM + k0 + kh;
            const float* w1 = W1 + (size_t)(nb0 + 16) * KDIM + k0 + kh;
            const float* w2 = W1 + (size_t)(nb0 + 32) * KDIM + k0 + kh;
            const float* w3 = W1 + (size_t)(nb0 + 48) * KDIM + k0 + kh;
            v2f bv;
            bv.x = w0[0]; bv.y = w0[1];
            acc0 = __builtin_amdgcn_wmma_f32_16x16x4_f32(false, a, false, bv, (short)0, acc0, false, false);
            bv.x = w1[0]; bv.y = w1[1];
            acc1 = __builtin_amdgcn_wmma_f32_16x16x4_f32(false, a, false, bv, (short)0, acc1, false, false);
            bv.x = w2[0]; bv.y = w2[1];
            acc2 = __builtin_amdgcn_wmma_f32_16x16x4_f32(false, a, false, bv, (short)0, acc2, false, false);
            bv.x = w3[0]; bv.y = w3[1];
            acc3 = __builtin_amdgcn_wmma_f32_16x16x4_f32(false, a, false, bv, (short)0, acc3, false, false);
        }
        #pragma unroll
        for (int tile = 0; tile < 4; ++tile) {
            const v8f acc = (tile == 0) ? acc0 : (tile == 1) ? acc1 : (tile == 2) ? acc2 : acc3;
            const int n = wave * 64 + tile * 16 + lr;
            const float bias = (float)DEG * b1[n];
            #pragma unroll
            for (int v = 0; v < 8; ++v)
                T[rbase + v][FDIM + n] = acc[v] + bias;
        }
    }
    __syncthreads();

    // ---- Phase 3: U = T @ W2^T + b2  -> S[:, 0:256] (S free now) ----
    {
        v8f acc0 = {}, acc1 = {}, acc2 = {}, acc3 = {};
        for (int k0 = 0; k0 < KDIM; k0 += 4) {
            v2f a;
            a.x = T[lr][k0 + kh];
            a.y = T[lr][k0 + kh + 1];
            const int nb0 = wave * 64 + lr;
            const float* w0 = W2 + (size_t)(nb0     ) * KDIM + k0 + kh;
            const float* w1 = W2 + (size_t)(nb0 + 16) * KDIM + k0 + kh;
            const float* w2 = W2 + (size_t)(nb0 + 32) * KDIM + k0 + kh;
            const float* w3 = W2 + (size_t)(nb0 + 48) * KDIM + k0 + kh;
            v2f bv;
            bv.x = w0[0]; bv.y = w0[1];
            acc0 = __builtin_amdgcn_wmma_f32_16x16x4_f32(false, a, false, bv, (short)0, acc0, false, false);
            bv.x = w1[0]; bv.y = w1[1];
            acc1 = __builtin_amdgcn_wmma_f32_16x16x4_f32(false, a, false, bv, (short)0, acc1, false, false);
            bv.x = w2[0]; bv.y = w2[1];
            acc2 = __builtin_amdgcn_wmma_f32_16x16x4_f32(false, a, false, bv, (short)0, acc2, false, false);
            bv.x = w3[0]; bv.y = w3[1];
            acc3 = __builtin_amdgcn_wmma_f32_16x16x4_f32(false, a, false, bv, (short)0, acc3, false, false);
        }
        #pragma unroll
        for (int tile = 0; tile < 4; ++tile) {
            const v8f acc = (tile == 0) ? acc0 : (tile == 1) ? acc1 : (tile == 2) ? acc2 : acc3;
            const int n = wave * 64 + tile * 16 + lr;
            const float bias = b2[n];
            #pragma unroll
            for (int v = 0; v < 8; ++v)
                S[rbase + v][n] = acc[v] + bias;    // U lives in S[:, 0:256]
        }
    }
    __syncthreads();

    // ---- Phase 4: row L2 norms + normalized write-out ----
    {
        const int r = t >> 3, seg = t & 7;
        float s = 0.f;
        #pragma unroll 8
        for (int c = seg * 32; c < seg * 32 + 32; ++c) {
            const float x = S[r][c];
            s += x * x;
        }
        part[r][seg] = s;
    }
    __syncthreads();
    if (t < TM) {
        float s = 0.f;
        #pragma unroll
        for (int i = 0; i < 8; ++i) s += part[t][i];
        inv_norm[t] = 1.0f / fmaxf(sqrtf(s), 1e-12f);
    }
    __syncthreads();
    for (int i = t; i < TM * FDIM; i += THREADS) {
        const int r = i >> 8;
        const int c = i & (FDIM - 1);
        const int g = m0 + r;
        if (g < nNodes)
            out[(size_t)g * FDIM + c] = S[r][c] * inv_norm[r];
    }
}

extern "C" void kernel_launch(void* const* d_in, const int* in_sizes, int n_in,
                              void* d_out, int out_size, void* d_ws, size_t ws_size,
                              hipStream_t stream) {
    const float* feat_data = (const float*)d_in[0];
    const float* feats     = (const float*)d_in[1];
    const float* W1        = (const float*)d_in[2];
    const float* b1        = (const float*)d_in[3];
    const float* W2        = (const float*)d_in[4];
    const float* b2        = (const float*)d_in[5];
    const int*   adj       = (const int*)d_in[6];
    const int*   in1       = (const int*)d_in[7];
    const int*   in2       = (const int*)d_in[8];
    const int*   neg       = (const int*)d_in[9];
    float* out = (float*)d_out;

    const int B1 = in_sizes[7];
    const int B2 = in_sizes[8];
    const int Bn = in_sizes[9];

    dim3 blk(THREADS);
    graphsage_fused_kernel<<<dim3((B1 + TM - 1) / TM), blk, 0, stream>>>(
        feat_data, feats, W1, b1, W2, b2, adj, in1, B1, out);
    graphsage_fused_kernel<<<dim3((B2 + TM - 1) / TM), blk, 0, stream>>>(
        feat_data, feats, W1, b1, W2, b2, adj, in2, B2, out + (size_t)B1 * FDIM);
    graphsage_fused_kernel<<<dim3((Bn + TM - 1) / TM), blk, 0, stream>>>(
        feat_data, feats, W1, b1, W2, b2, adj, neg, Bn, out + (size_t)(B1 + B2) * FDIM);
}